// Entmax15_17197049053909
// MI455X (gfx1250) — compile-verified
//
#include <hip/hip_runtime.h>

// Entmax-1.5 (15 fixed-point iterations) over rows of a 4096 x 32000 fp32 matrix.
// One workgroup (1024 threads = 32 wave32 waves) per row. Each thread owns
// 8 float4 chunks (COLS = 32000 = 8000 B128 chunks exactly); all iteration
// state lives in VGPRs (32 floats/thread). The row is staged into LDS once via
// CDNA5 async global->LDS B128 copies (ASYNCcnt), then gathered with
// ds_load_b128. Reductions: ds_swizzle XOR within each wave32 + tiny LDS
// scratch across the 32 waves. Output: B128 non-temporal stores.

#define COLS   32000
#define BLOCK  1024
#define NCHUNK 8000        // COLS*4/16 : B128 chunks per row (exact)
#define JMAX   8           // chunks per thread (8*1024 = 8192 >= 8000)
#define TAILT  832         // NCHUNK - 7*1024 : threads with a valid 8th chunk
#define KMAX   32          // floats per thread
#define N_ITER 15
#define EPS    1e-12f
#define L2E    1.4426950408889634f

typedef float v4f __attribute__((ext_vector_type(4)));   // native vec: B128 ops

// ---------- wave32 XOR shuffle via ds_swizzle (imm must be a constant) ----------
template <int M>
__device__ __forceinline__ float xswap(float v) {
    int i = __builtin_bit_cast(int, v);
    i = __builtin_amdgcn_ds_swizzle(i, 0x1F | (M << 10));   // and=0x1F, or=0, xor=M
    return __builtin_bit_cast(float, i);
}

__device__ __forceinline__ float waveSum(float v) {
    v += xswap<1>(v); v += xswap<2>(v); v += xswap<4>(v);
    v += xswap<8>(v); v += xswap<16>(v);
    return v;
}
__device__ __forceinline__ float waveMax(float v) {
    v = fmaxf(v, xswap<1>(v)); v = fmaxf(v, xswap<2>(v)); v = fmaxf(v, xswap<4>(v));
    v = fmaxf(v, xswap<8>(v)); v = fmaxf(v, xswap<16>(v));
    return v;
}

// ---------- block-wide reductions (32 waves), red[] = 66 floats ----------
__device__ __forceinline__ void blockSum2(float& a, float& b, float* red, int tid) {
    a = waveSum(a); b = waveSum(b);
    const int w = tid >> 5, l = tid & 31;
    if (l == 0) { red[w] = a; red[32 + w] = b; }
    __syncthreads();
    if (w == 0) {
        float x = waveSum(red[l]);
        float y = waveSum(red[32 + l]);
        if (l == 0) { red[64] = x; red[65] = y; }
    }
    __syncthreads();
    a = red[64]; b = red[65];
}

__device__ __forceinline__ float blockSum1(float a, float* red, int tid) {
    a = waveSum(a);
    const int w = tid >> 5, l = tid & 31;
    if (l == 0) red[w] = a;
    __syncthreads();
    if (w == 0) {
        float x = waveSum(red[l]);
        if (l == 0) red[64] = x;
    }
    __syncthreads();
    return red[64];
}

__device__ __forceinline__ float blockMax(float a, float* red, int tid) {
    a = waveMax(a);
    const int w = tid >> 5, l = tid & 31;
    if (l == 0) red[w] = a;
    __syncthreads();
    if (w == 0) {
        float x = waveMax(red[l]);
        if (l == 0) red[64] = x;
    }
    __syncthreads();
    return red[64];
}

// ---------- fast math (raw trans instructions; guarded fallbacks) ----------
__device__ __forceinline__ float fast_rcp(float x) {
#if __has_builtin(__builtin_amdgcn_rcpf)
    return __builtin_amdgcn_rcpf(x);
#else
    return 1.0f / x;
#endif
}
__device__ __forceinline__ float fast_rsq(float x) {
#if __has_builtin(__builtin_amdgcn_rsqf)
    return __builtin_amdgcn_rsqf(x);
#else
    return rsqrtf(x);
#endif
}
__device__ __forceinline__ float fast_exp2(float x) {
#if __has_builtin(__builtin_amdgcn_exp2f)
    return __builtin_amdgcn_exp2f(x);
#else
    return exp2f(x);
#endif
}

__global__ __launch_bounds__(BLOCK)
void entmax15_rows(const float* __restrict__ logits, float* __restrict__ out) {
    extern __shared__ float row[];      // COLS floats = 128000 B dynamic LDS
    __shared__ float red[66];
    const int tid = (int)threadIdx.x;
    const float* g = logits + (long long)blockIdx.x * COLS;
    const bool ok7 = tid < TAILT;       // thread's 8th chunk is valid

    // ---- stage the row into LDS with CDNA5 async B128 copies (ASYNCcnt) ----
    {
        const unsigned ldsBase = (unsigned)(size_t)row;  // LDS byte offset
        #pragma unroll
        for (int j = 0; j < JMAX; ++j) {
            const int c = tid + j * BLOCK;
            if (j < JMAX - 1 || ok7) {
                const unsigned loff = ldsBase + (unsigned)c * 16u;
                const float* gp = g + (size_t)c * 4;
                asm volatile("global_load_async_to_lds_b128 %0, %1, off"
                             :: "v"(loff), "v"(gp) : "memory");
            }
        }
        asm volatile("s_wait_asynccnt 0" ::: "memory");
    }
    __syncthreads();

    // ---- gather this thread's 8 float4 chunks from LDS (ds_load_b128) ----
    float a[KMAX];
    {
        const v4f* row4 = (const v4f*)row;
        #pragma unroll
        for (int j = 0; j < JMAX; ++j) {
            const int c = tid + j * BLOCK;
            if (j < JMAX - 1 || ok7) {
                const v4f v = row4[c];
                a[4 * j + 0] = v.x; a[4 * j + 1] = v.y;
                a[4 * j + 2] = v.z; a[4 * j + 3] = v.w;
            } else {
                a[4 * j + 0] = a[4 * j + 1] = a[4 * j + 2] = a[4 * j + 3] =
                    -__builtin_huge_valf();
            }
        }
    }

    // ---- softmax ----
    float m = -__builtin_huge_valf();
    #pragma unroll
    for (int k = 0; k < KMAX; ++k) m = fmaxf(m, a[k]);
    m = blockMax(m, red, tid);

    float s0 = 0.0f;
    #pragma unroll
    for (int k = 0; k < KMAX; ++k) {
        const float e = fast_exp2((a[k] - m) * L2E);  // invalid slot: exp(-inf)=0
        a[k] = e;
        s0 += e;
    }
    s0 = blockSum1(s0, red, tid);
    float zinv = fast_rcp(s0);          // current alpha[k] == a[k] * zinv

    // ---- 15 fixed-point iterations, fully register-resident ----
    #pragma unroll 1
    for (int it = 0; it < N_ITER; ++it) {
        float ssum = 0.0f, sinv = 0.0f;
        #pragma unroll
        for (int k = 0; k < KMAX; ++k) {
            const float ac  = fmaxf(a[k] * zinv, EPS);   // clip(alpha, EPS)
            const float rq  = fast_rsq(ac);              // 1/sqrt(alpha)
            const float asq = ac * rq;                   // sqrt(alpha)
            if (k < 28 || ok7) { ssum += asq; sinv += rq; }
            a[k] = asq;
        }
        blockSum2(ssum, sinv, red, tid);
        const float tau = (ssum - 1.0f) * fast_rcp(sinv);

        float z = 0.0f;
        #pragma unroll
        for (int k = 0; k < KMAX; ++k) {
            const float t  = a[k] + tau;
            const float an = t * t;                      // clip(.,0) free: square
            if (k < 28 || ok7) z += an;
            a[k] = an;
        }
        z = blockSum1(z, red, tid);
        zinv = fast_rcp(z);
    }

    // ---- store normalized result: B128 non-temporal, coalesced ----
    v4f* o4 = (v4f*)(out + (long long)blockIdx.x * COLS);
    #pragma unroll
    for (int j = 0; j < JMAX; ++j) {
        const int c = tid + j * BLOCK;
        if (j < JMAX - 1 || ok7) {
            v4f v;
            v.x = a[4 * j + 0] * zinv; v.y = a[4 * j + 1] * zinv;
            v.z = a[4 * j + 2] * zinv; v.w = a[4 * j + 3] * zinv;
            __builtin_nontemporal_store(v, &o4[c]);
        }
    }
}

extern "C" void kernel_launch(void* const* d_in, const int* in_sizes, int n_in,
                              void* d_out, int out_size, void* d_ws, size_t ws_size,
                              hipStream_t stream) {
    const float* logits = (const float*)d_in[0];
    float* out = (float*)d_out;
    const int rows = in_sizes[0] / COLS;               // 4096
    const size_t shmem = (size_t)COLS * sizeof(float); // 128000 B
    entmax15_rows<<<dim3((unsigned)rows), dim3(BLOCK), shmem, stream>>>(logits, out);
    (void)n_in; (void)out_size; (void)d_ws; (void)ws_size;
}